// YOLOv3Layer_15290083573987
// MI455X (gfx1250) — compile-verified
//
#include <hip/hip_runtime.h>
#include <stdint.h>

#define C_TOT 255          // 3 boxes * 85
#define BOXD  85
#define HDIM  52
#define WDIM  52
#define HW    (HDIM * WDIM)            // 2704
#define TILE_FLOATS (C_TOT * WDIM)     // 13260 floats = 53040 B LDS
#define NQUAD (C_TOT * 13)             // 3315 float4 chunks per tile

__global__ __launch_bounds__(256)
void yolo_head_kernel(const float* __restrict__ in, float* __restrict__ out, int B) {
    __shared__ float  tile[TILE_FLOATS];   // [c][52], pitch 208 B (16B aligned)
    __shared__ float4 ctab[C_TOT];         // per-channel {den_off, w_coeff, add_const, scale}

    const int tid = threadIdx.x;
    const int blk = blockIdx.x;
    const int b = blk / HDIM;
    const int h = blk - b * HDIM;

    // trailing scalar output (-1.0f) after the big tensor
    if (blk == 0 && tid == 0) {
        out[(size_t)B * HW * 3 * BOXD] = -1.0f;
    }

    // ------------- Phase 1a: issue async global->LDS DMA (coalesced b128 per row) ------
    const float* gbase = in + ((size_t)b * C_TOT) * HW + (size_t)h * WDIM;
    const uint32_t lds_base = (uint32_t)(uintptr_t)tile;   // low 32 bits = LDS offset

    for (int q = tid; q < NQUAD; q += 256) {
        int c = q / 13;            // channel 0..254
        int j = q - c * 13;        // float4 chunk within row, 0..12
        uint32_t voff  = (uint32_t)((c * HW + j * 4) * 4);        // byte offset from gbase
        uint32_t laddr = lds_base + (uint32_t)((c * WDIM + j * 4) * 4);
        // GVS mode: mem = SADDR(64b sgpr) + VADDR(32b vgpr) ; dest LDS addr in VDST vgpr
        asm volatile("global_load_async_to_lds_b128 %0, %1, %2"
                     :: "v"(laddr), "v"(voff), "s"(gbase)
                     : "memory");
    }

    // ------------- Phase 1b: build per-channel table while DMA is in flight ------------
    // d==0: (sig+w)/52  d==1: (sig+h)/52  d==2: exp*aw/416  d==3: exp*ah/416  else: sig
    if (tid < C_TOT) {
        int c   = tid;
        int box = c / BOXD;
        int d   = c - box * BOXD;
        const float AW[3] = {10.0f / 416.0f, 16.0f / 416.0f, 33.0f / 416.0f};
        const float AH[3] = {13.0f / 416.0f, 30.0f / 416.0f, 23.0f / 416.0f};
        float4 k;
        k.x = (d == 2 || d == 3) ? 0.0f : 1.0f;                       // den = E + k.x
        k.y = (d == 0) ? 1.0f : 0.0f;                                 // * w
        k.z = (d == 1) ? (float)h : 0.0f;                             // + h
        k.w = (d < 2) ? (1.0f / 52.0f)
                      : ((d == 2) ? AW[box] : ((d == 3) ? AH[box] : 1.0f));
        ctab[c] = k;
    }

#if __has_builtin(__builtin_amdgcn_s_wait_asynccnt)
    __builtin_amdgcn_s_wait_asynccnt(0);
#else
    asm volatile("s_wait_asynccnt 0" ::: "memory");
#endif
    __syncthreads();

    // ------------- Phase 2: elementwise math, perfectly coalesced stores ---------------
    // out flat offset = b*689520 + h*13260 + e, where e = w*255 + c
    const float LOG2E = 1.4426950408889634f;
    float* obase = out + (size_t)b * (HW * 3 * BOXD) + (size_t)h * TILE_FLOATS;

    int w = tid / C_TOT;           // tid < 256  =>  w in {0,1}
    int c = tid - w * C_TOT;

    for (int e = tid; e < TILE_FLOATS; e += 256) {
        float  v = tile[c * WDIM + w];
        float4 k = ctab[c];

        float E = __builtin_amdgcn_exp2f(v * -LOG2E);        // exp(-v)
        float t = __builtin_amdgcn_rcpf(E + k.x);            // exp(v) or sigmoid(v)
        float r = (t + __builtin_fmaf(k.y, (float)w, k.z)) * k.w;

        obase[e] = r;

        // advance e by 256 = 255 + 1  ->  c += 1 (wrap 255), w += 1 (+carry)
        c += 1; w += 1;
        if (c == C_TOT) { c = 0; w += 1; }
    }
}

extern "C" void kernel_launch(void* const* d_in, const int* in_sizes, int n_in,
                              void* d_out, int out_size, void* d_ws, size_t ws_size,
                              hipStream_t stream) {
    (void)n_in; (void)d_ws; (void)ws_size; (void)out_size;
    const float* x = (const float*)d_in[0];
    float* out = (float*)d_out;
    int B = in_sizes[0] / (C_TOT * HW);   // 64 for the reference shapes
    dim3 grid(B * HDIM);                  // one block per (batch, image row): 3328
    dim3 block(256);
    yolo_head_kernel<<<grid, block, 0, stream>>>(x, out, B);
}